// LLADASampler_54082228191627
// MI455X (gfx1250) — compile-verified
//
#include <hip/hip_runtime.h>
#include <cstdint>

#define MASK_INDEX 1
#define NB 4
#define NS 1024
#define NV 32000
#define BLK_A 320                       // 10 wave32 waves
#define F4_PER_ROW (NV / 4)             // 8000 float4 per row
#define ITERS_A (F4_PER_ROW / BLK_A)    // 25, exact

__device__ __forceinline__ void async_cp16(uint32_t lds_off, uint64_t gaddr) {
  // CDNA5 async data mover: global -> LDS, per-lane, tracked by ASYNCcnt.
  asm volatile("global_load_async_to_lds_b128 %0, %1, off"
               :: "v"(lds_off), "v"(gaddr) : "memory");
}

__device__ __forceinline__ void upd_max(float sc, int idx, float l,
                                        float& best, int& bidx, float& bl) {
  if (sc > best) { best = sc; bidx = idx; bl = l; }   // strict > keeps first index
}

// Streams one vocab row through a depth-2 async global->LDS pipeline.
// Each lane owns its LDS slot, so no workgroup barriers are needed:
// ASYNCcnt orders fills (in-order completion), DScnt guards buffer reuse.
template <bool WITH_SUM>
__device__ __forceinline__ void scan_row_async(
    const float4* __restrict__ lp, const float4* __restrict__ up, int tid,
    float4 (&sbuf)[2][2][BLK_A],
    float& best, int& bidx, float& bl, float& sum) {
  const uint32_t dL[2] = {(uint32_t)(uintptr_t)&sbuf[0][0][tid],
                          (uint32_t)(uintptr_t)&sbuf[1][0][tid]};
  const uint32_t dU[2] = {(uint32_t)(uintptr_t)&sbuf[0][1][tid],
                          (uint32_t)(uintptr_t)&sbuf[1][1][tid]};
  uint64_t aL = (uint64_t)(uintptr_t)(lp + tid);
  uint64_t aU = (uint64_t)(uintptr_t)(up + tid);
  const uint64_t step = (uint64_t)BLK_A * sizeof(float4);   // 5120 B

  // Prime both buffers (iters 0 and 1): 4 async ops outstanding.
  async_cp16(dL[0], aL); async_cp16(dU[0], aU); aL += step; aU += step;
  async_cp16(dL[1], aL); async_cp16(dU[1], aU); aL += step; aU += step;

  for (int i = 0; i < ITERS_A; ++i) {
    if (i + 1 < ITERS_A)
      asm volatile("s_wait_asynccnt 0x2" ::: "memory");  // iter i's fills done
    else
      asm volatile("s_wait_asynccnt 0x0" ::: "memory");  // drain at tail
    const int cb = i & 1;
    float4 l4 = sbuf[cb][0][tid];                        // ds_load_b128 (own slot)
    float4 u4 = sbuf[cb][1][tid];
    asm volatile("s_wait_dscnt 0x0" ::: "memory");       // data in VGPRs -> slot reusable
    if (i + 2 < ITERS_A) {                               // re-arm this buffer
      async_cp16(dL[cb], aL);
      async_cp16(dU[cb], aU);
      aL += step; aU += step;
    }
    const int e0 = (i * BLK_A + tid) * 4;
    float ls[4] = {l4.x, l4.y, l4.z, l4.w};
    float us[4] = {u4.x, u4.y, u4.z, u4.w};
#pragma unroll
    for (int k = 0; k < 4; ++k) {
      // score = l/T - log(-log(u))  (monotone transform of exp(l/T)/(-log u))
      float g  = -__logf(us[k]);                                   // > 0
      float sc = __builtin_fmaf(ls[k], (1.0f / 3.0f), -__logf(g));
      upd_max(sc, e0 + k, ls[k], best, bidx, bl);
      if (WITH_SUM) sum += __expf(ls[k]);                          // softmax denom
    }
  }
}

// One block per (b,s) row: argmax + softmax-prob of sampled token + masking.
__global__ __launch_bounds__(BLK_A) void llada_row_kernel(
    const float* __restrict__ logits, const float* __restrict__ noise,
    const int* __restrict__ x_t, const int* __restrict__ plen_ptr,
    float* __restrict__ conf_out, int* __restrict__ x0_ws) {
  const int row  = blockIdx.x;            // 0..NB*NS-1
  const int s    = row & (NS - 1);
  const int tid  = threadIdx.x;
  const int plen = plen_ptr[0];
  const long base = (long)row * NV;

  __shared__ float4 sbuf[2][2][BLK_A];    // [buffer][logits|noise][lane slot]

  float best = -__builtin_inff();
  int   bidx = 0;
  float bl = 0.0f, sum = 0.0f;
  const float4* lp = reinterpret_cast<const float4*>(logits + base);
  const float4* up = reinterpret_cast<const float4*>(noise + base);
  if (s < plen)  // uniform per block: only prompt columns ever need the softmax sum
    scan_row_async<true >(lp, up, tid, sbuf, best, bidx, bl, sum);
  else
    scan_row_async<false>(lp, up, tid, sbuf, best, bidx, bl, sum);

  // wave32 butterfly reduce (argmax with low-index tiebreak, plus sum)
#pragma unroll
  for (int off = 16; off > 0; off >>= 1) {
    float ob = __shfl_xor(best, off, 32);
    int   oi = __shfl_xor(bidx, off, 32);
    float ol = __shfl_xor(bl,   off, 32);
    float os = __shfl_xor(sum,  off, 32);
    sum += os;
    if (ob > best || (ob == best && oi < bidx)) { best = ob; bidx = oi; bl = ol; }
  }

  __shared__ float s_best[BLK_A / 32];
  __shared__ int   s_idx [BLK_A / 32];
  __shared__ float s_bl  [BLK_A / 32];
  __shared__ float s_sum [BLK_A / 32];
  if ((tid & 31) == 0) {
    const int w = tid >> 5;
    s_best[w] = best; s_idx[w] = bidx; s_bl[w] = bl; s_sum[w] = sum;
  }
  __syncthreads();

  if (tid == 0) {
    best = s_best[0]; bidx = s_idx[0]; bl = s_bl[0]; sum = s_sum[0];
#pragma unroll
    for (int w = 1; w < BLK_A / 32; ++w) {
      const float ob = s_best[w];
      const int   oi = s_idx[w];
      sum += s_sum[w];
      if (ob > best || (ob == best && oi < bidx)) { best = ob; bidx = oi; bl = s_bl[w]; }
    }
    const int  xt      = x_t[row];
    const bool is_mask = (xt == MASK_INDEX);
    const float ninf   = -__builtin_inff();
    const float conf_p = (s < plen) ? (__expf(bl) / sum) : ninf;  // col mask first
    conf_out[row] = is_mask ? conf_p : ninf;
    x0_ws[row]    = is_mask ? bidx : xt;
  }
}

// One block per batch row: async-stage confidence row into LDS, rank-count top-k.
__global__ __launch_bounds__(NS) void llada_topk_kernel(
    const float* __restrict__ conf, const int* __restrict__ x_t,
    const int* __restrict__ ntt, const int* __restrict__ x0_ws,
    float* __restrict__ x0_out) {
  const int b   = blockIdx.x;
  const int tid = threadIdx.x;
  __shared__ float sc[NS];

  if (tid < NS / 4) {
    uint64_t src = (uint64_t)(uintptr_t)(conf + (long)b * NS + tid * 4);
    uint32_t dst = (uint32_t)(uintptr_t)(&sc[tid * 4]);   // low 32b = LDS offset
    async_cp16(dst, src);
  }
  asm volatile("s_wait_asynccnt 0x0" ::: "memory");
  __syncthreads();

  const float ci = sc[tid];
  int cnt = 0;  // rank = #{c_j > c_i} + #{j < i : c_j == c_i}  (stable desc sort)
  const float4* sc4 = reinterpret_cast<const float4*>(sc);
#pragma unroll 4
  for (int j4 = 0; j4 < NS / 4; ++j4) {
    const float4 c4 = sc4[j4];           // same addr across lanes -> LDS broadcast
    const int j = j4 * 4;
    cnt += (int)((c4.x > ci) | ((c4.x == ci) & (j + 0 < tid)));
    cnt += (int)((c4.y > ci) | ((c4.y == ci) & (j + 1 < tid)));
    cnt += (int)((c4.z > ci) | ((c4.z == ci) & (j + 2 < tid)));
    cnt += (int)((c4.w > ci) | ((c4.w == ci) & (j + 3 < tid)));
  }

  const int k   = ntt[b];
  const int idx = b * NS + tid;
  const int xt  = x_t[idx];
  const int x0  = (cnt < k) ? x0_ws[idx] : xt;
  x0_out[idx] = (float)x0;
}

extern "C" void kernel_launch(void* const* d_in, const int* in_sizes, int n_in,
                              void* d_out, int out_size, void* d_ws, size_t ws_size,
                              hipStream_t stream) {
  (void)in_sizes; (void)n_in; (void)out_size; (void)ws_size;
  const float* logits = (const float*)d_in[0];   // [B,S,V] f32
  const float* noise  = (const float*)d_in[1];   // [B,S,V] f32
  const int*   x_t    = (const int*)d_in[2];     // [B,S]  i32
  const int*   ntt    = (const int*)d_in[3];     // [B]    i32
  const int*   plen   = (const int*)d_in[4];     // scalar i32

  float* out      = (float*)d_out;               // [x0_new | confidence], 8192 f32
  float* conf_out = out + NB * NS;
  int*   x0_ws    = (int*)d_ws;                  // 4096 ints of scratch

  llada_row_kernel<<<NB * NS, BLK_A, 0, stream>>>(logits, noise, x_t, plen,
                                                  conf_out, x0_ws);
  llada_topk_kernel<<<NB, NS, 0, stream>>>(conf_out, x_t, ntt, x0_ws, out);
}